// GraphLSTM_39247411151129
// MI455X (gfx1250) — compile-verified
//
#include <hip/hip_runtime.h>
#include <hip/hip_bf16.h>
#include <math.h>

// Problem constants (from reference): M=N=32 -> MN=1024, D=8, H=64, 4H=256, O=1, B=8
#define MN    1024
#define BATCH 8
#define DIM   8
#define HH    64
#define G4    256           // 4*H
#define NWAVE 8             // waves per block

typedef float v2f __attribute__((ext_vector_type(2)));
typedef float v8f __attribute__((ext_vector_type(8)));

// Branch-free fast activations on v_exp_f32 / v_rcp_f32 (no libm tanhf blowup).
__device__ __forceinline__ float fast_sigmoid(float x) {
    return __builtin_amdgcn_rcpf(1.0f + __expf(-x));
}
__device__ __forceinline__ float fast_tanh(float x) {
    return 2.0f * __builtin_amdgcn_rcpf(1.0f + __expf(-2.0f * x)) - 1.0f;
}

// One block per destination node. dst_idx layout (from reference graph build):
//   e in [0, MN)  : self edges, dst = e
//   e in [MN, E)  : nonzero edges sorted ascending by dst (numpy row-major nonzero)
// Block nodeId processes {self edge nodeId} + contiguous range [lo, hi).
//
// Per edge, one wave computes gates via V_WMMA_F32_16X16X4_F32 (two K=4 steps):
//   A 16x4 f32: lane l: m = l&15 (batch row, rows 8..15 zero), vgpr v: K = 2*(l>>4)+v
//   B 4x16 f32: lane l: n = l&15,  vgpr v: K = 2*(l>>4)+v  (W_ih[e,g,:] rows are 8 f32)
//   D 16x16:    lanes 0..15 = gate col, vgpr j = batch row j
// The f-gate block (cols 64..127 of W_ih/b_ih/b_hh) is dead in the reference ->
// never loaded, cutting streamed bytes by 25%.
__global__ __launch_bounds__(256) void graph_lstm_fused(
    const float* __restrict__ x,       // (8, 8, 1024)
    const float* __restrict__ edge_w,  // (E)
    const float* __restrict__ W_ih,    // (E, 256, 8)
    const float* __restrict__ b_ih,    // (E, 256)
    const float* __restrict__ b_hh,    // (E, 256)
    const float* __restrict__ conv_w,  // (1, 64)
    const float* __restrict__ conv_b,  // (1,)
    const int*   __restrict__ src_idx, // (E)
    const int*   __restrict__ dst_idx, // (E)
    float*       __restrict__ out,     // (8, 1, 32, 32) = b*1024 + mn
    int E)
{
    __shared__ float partial[NWAVE * BATCH * HH]; // 8 waves x 512 = 16 KB
    __shared__ float reduced[BATCH * HH];         // 512 floats

    const int nodeId = blockIdx.x;
    const int lane   = threadIdx.x & 31;
    const int wave   = threadIdx.x >> 5;
    const int tid    = threadIdx.x;

    // ---- binary search the sorted dst range [lo, hi) for this node ----
    int a = MN, b = E;
    while (a < b) { int mid = (a + b) >> 1; if (dst_idx[mid] < nodeId) a = mid + 1; else b = mid; }
    const int lo = a;
    b = E;
    while (a < b) { int mid = (a + b) >> 1; if (dst_idx[mid] <= nodeId) a = mid + 1; else b = mid; }
    const int hi  = a;
    const int cnt = 1 + (hi - lo);     // + self edge

    const int m  = lane & 15;          // batch row (A) / gate col (B,D)
    const int kb = (lane >> 4) << 1;   // K base for this lane half

    // Per-wave accumulators in D layout: acc[hc][j] = sum over edges of
    // edge_w * h_new  at (b=j, h=hc*16 + lane) for lanes 0..15.
    v8f acc0 = {}, acc1 = {}, acc2 = {}, acc3 = {};
    v8f* accs[4] = { &acc0, &acc1, &acc2, &acc3 };

    for (int t = wave; t < cnt; t += NWAVE) {      // wave-uniform loop bound
        const int   e   = (t == 0) ? nodeId : (lo + t - 1);
        const int   src = src_idx[e];
        const float ew  = edge_w[e];

        // ---- A fragments: a0 -> d = kb,kb+1 ; a1 -> d = kb+4,kb+5 ----
        v2f a0 = (v2f)(0.0f), a1 = (v2f)(0.0f);
        if (m < BATCH) {
            const float* xb = x + (size_t)m * (DIM * MN) + src;
            a0.x = xb[(size_t)(kb + 0) * MN];
            a0.y = xb[(size_t)(kb + 1) * MN];
            a1.x = xb[(size_t)(kb + 4) * MN];
            a1.y = xb[(size_t)(kb + 5) * MN];
        }

        const float* We = W_ih + (size_t)e * (G4 * DIM);
        const float* Bi = b_ih + (size_t)e * G4;
        const float* Bh = b_hh + (size_t)e * G4;

        auto tile = [&](int gcol) -> v8f {
            const float* wr = We + (size_t)gcol * DIM;
            v2f bA = *(const v2f*)(wr + kb);      // K = kb, kb+1
            v2f bB = *(const v2f*)(wr + kb + 4);  // K = kb+4, kb+5
            v8f c = {};
            c = __builtin_amdgcn_wmma_f32_16x16x4_f32(false, a0, false, bA,
                                                      (short)0, c, false, false);
            c = __builtin_amdgcn_wmma_f32_16x16x4_f32(false, a1, false, bB,
                                                      (short)0, c, false, false);
            return c;
        };

        #pragma unroll
        for (int hc = 0; hc < 4; ++hc) {
            const int gi = hc * 16 + m;

            v8f ci = tile(gi);            // i-gate
            v8f cg = tile(128 + gi);      // g-gate
            v8f co = tile(192 + gi);      // o-gate

            const float bi = Bi[gi]       + Bh[gi];
            const float bg = Bi[128 + gi] + Bh[128 + gi];
            const float bo = Bi[192 + gi] + Bh[192 + gi];

            v8f av = *accs[hc];
            #pragma unroll
            for (int j = 0; j < BATCH; ++j) {     // all lanes compute; lanes
                const float igate = ci[j] + bi;   // 16..31 produce don't-care
                const float ggate = cg[j] + bg;   // values discarded below
                const float ogate = co[j] + bo;
                const float c_new = fast_sigmoid(igate) * fast_tanh(ggate);
                const float h_new = fast_sigmoid(ogate) * fast_tanh(c_new);
                av[j] += ew * h_new;
            }
            *accs[hc] = av;
        }
    }

    // ---- dump per-wave partials to LDS: partial[wave][b=j][h=hc*16+lane] ----
    if (lane < 16) {
        float* pw = partial + wave * (BATCH * HH);
        #pragma unroll
        for (int hc = 0; hc < 4; ++hc) {
            v8f av = *accs[hc];
            #pragma unroll
            for (int j = 0; j < BATCH; ++j)
                pw[j * HH + hc * 16 + lane] = av[j];
        }
    }
    __syncthreads();

    // ---- cross-wave tree sum: 512 outputs, thread t handles t and t+256 ----
    {
        float s0 = 0.0f, s1 = 0.0f;
        #pragma unroll
        for (int w = 0; w < NWAVE; ++w) {
            s0 += partial[w * (BATCH * HH) + tid];
            s1 += partial[w * (BATCH * HH) + 256 + tid];
        }
        reduced[tid]       = s0;
        reduced[tid + 256] = s1;
    }
    __syncthreads();

    // ---- fused conv: wave w computes out[b=w, nodeId] = <node[w,:], conv_w> + cb ----
    {
        const float* nb = reduced + wave * HH;
        float v = nb[lane]      * conv_w[lane]
                + nb[lane + 32] * conv_w[lane + 32];
        #pragma unroll
        for (int off = 16; off > 0; off >>= 1)
            v += __shfl_xor(v, off, 32);
        if (lane == 0)
            out[wave * MN + nodeId] = v + conv_b[0];
    }
}

extern "C" void kernel_launch(void* const* d_in, const int* in_sizes, int n_in,
                              void* d_out, int out_size, void* d_ws, size_t ws_size,
                              hipStream_t stream) {
    const float* x      = (const float*)d_in[0];
    const float* edge_w = (const float*)d_in[1];
    const float* W_ih   = (const float*)d_in[2];
    const float* b_ih   = (const float*)d_in[3];
    const float* b_hh   = (const float*)d_in[4];
    const float* conv_w = (const float*)d_in[5];
    const float* conv_b = (const float*)d_in[6];
    const int*   src_i  = (const int*)d_in[7];
    const int*   dst_i  = (const int*)d_in[8];
    float*       out    = (float*)d_out;
    const int E = in_sizes[1];                   // edge_w element count

    graph_lstm_fused<<<MN, NWAVE * 32, 0, stream>>>(
        x, edge_w, W_ih, b_ih, b_hh, conv_w, conv_b, src_i, dst_i, out, E);
}